// PhiNN_45483703664653
// MI455X (gfx1250) — compile-verified
//
#include <hip/hip_runtime.h>
#include <math.h>

// ---------------------------------------------------------------------------
// Fused SDE integrator for the per-particle MLP potential phi.
// Each wave (32 lanes) owns 16 particles for all nsteps. MLP fwd+bwd done with
// V_WMMA_F32_16X16X4_F32 (fp32 tensor op, 16x16 tiles, K chained by 4).
// ---------------------------------------------------------------------------

typedef float v2f __attribute__((ext_vector_type(2)));
typedef float v8f __attribute__((ext_vector_type(8)));

#define WAVES 4
#define NB    128          // B
#define NP    2048         // N
#define ROW   (2 + NP*2 + 5)   // 4103 floats per x row
#define SIGMA_C 0.001f

__device__ __forceinline__ v8f wmma4(v2f a, v2f b, v8f c) {
  return __builtin_amdgcn_wmma_f32_16x16x4_f32(false, a, false, b, (short)0, c,
                                               false, false);
}

__device__ __forceinline__ float eluf(float v) {
  return v > 0.f ? v : expm1f(v);
}
__device__ __forceinline__ float elup_from_h(float h) {  // elu'(z) given h=elu(z)
  return h > 0.f ? 1.f : h + 1.f;
}

// splitmix64-based counter RNG -> standard normal (Box-Muller)
__device__ __forceinline__ float rng_normal(unsigned step, unsigned idx) {
  unsigned long long z =
      (((unsigned long long)step << 32) ^ (unsigned long long)idx) *
          0x9E3779B97F4A7C15ull +
      0x2545F4914F6CDD1Dull;
  z ^= z >> 30; z *= 0xBF58476D1CE4E5B9ull;
  z ^= z >> 27; z *= 0x94D049BB133111EBull;
  z ^= z >> 31;
  unsigned u1 = (unsigned)z, u2 = (unsigned)(z >> 32);
  float a = (float)(u1 >> 8) * (1.0f / 16777216.0f) + (0.5f / 16777216.0f);
  float b = (float)(u2 >> 8) * (1.0f / 16777216.0f);
  return sqrtf(-2.f * __logf(a)) * __cosf(6.28318530718f * b);
}

struct Slab {
  float yb[32];        // input tile 16p x 2 (y)
  float h1[16 * 16];   // activations per layer, [particle][feature]
  float h2[16 * 32];
  float h3[16 * 32];
  float h4[16 * 16];
  float g5[16];        // sigmoid(z5) per particle
  float gA[16 * 32];   // gradient ping
  float gB[16 * 32];   // gradient pong
};

// A-fragment: 16xK tile at chunk c from LDS [particle][feature] (stride S)
__device__ __forceinline__ v2f afrag(const float* t, int q, int half, int c,
                                     int S) {
  int k = c * 4 + 2 * half;
  v2f a; a.x = t[q * S + k]; a.y = t[q * S + k + 1];
  return a;
}

__global__ __launch_bounds__(WAVES * 32, 1) void sde_mlp_kernel(
    const float* __restrict__ x,
    const float* __restrict__ W1, const float* __restrict__ b1,
    const float* __restrict__ W2, const float* __restrict__ b2,
    const float* __restrict__ W3, const float* __restrict__ b3,
    const float* __restrict__ W4, const float* __restrict__ b4,
    const float* __restrict__ W5, const float* __restrict__ b5,
    const float* __restrict__ Wt, const float* __restrict__ dtp,
    const int* __restrict__ nsp, float* __restrict__ out) {
  __shared__ Slab slabs[WAVES];
  __shared__ float sW1[32];  // W1 (16x2) row-major
  __shared__ float sW5[16];  // W5 (1x16)

  const int tid = threadIdx.x;
  const int wid = tid >> 5;
  const int lane = tid & 31;
  const int q = lane & 15;     // particle (A/D row) or feature (B/D col)
  const int half = lane >> 4;  // lane half selects K / M sub-blocks

  if (tid < 32) sW1[tid] = W1[tid];
  else if (tid < 48) sW5[tid - 32] = W5[tid - 32];
  __syncthreads();

  Slab& S = slabs[wid];

  const int tile = blockIdx.x * WAVES + wid;   // 0..16383
  const int b = tile >> 7;                     // 128 tiles per batch row
  const int n0 = (tile & 127) << 4;            // first particle of tile

  const float* xr = x + (size_t)b * ROW;
  const float t0 = xr[0];
  const float tcrit = xr[2 + NP * 2 + 0];
  const float sp0 = xr[2 + NP * 2 + 1], sp1 = xr[2 + NP * 2 + 2];
  const float ss0 = xr[2 + NP * 2 + 3], ss1 = xr[2 + NP * 2 + 4];
  const float wt0 = Wt[half * 2 + 0], wt1 = Wt[half * 2 + 1]; // row d=half
  const float dtv = dtp[0];
  const int nsteps = nsp[0];
  const float sqdt = sqrtf(dtv);

  // this lane owns y[particle=q][dim=half]
  float yown = xr[2 + (n0 + q) * 2 + half];
  const unsigned gidx = (unsigned)(b * NP * 2 + (n0 + q) * 2 + half);

  // ---------------- preload weight B-fragments (registers, reused 50x) ------
  // forward: B = W^T (K = in-dim rows, N = out-dim cols); W row-major (out,in)
  v2f f1;  // layer1: K=2 padded to 4 (rows 2,3 zero)
  f1.x = half ? 0.f : W1[2 * q + 0];
  f1.y = half ? 0.f : W1[2 * q + 1];

  v2f f2[2][4], f3[2][8], f4[8];
  v2f r4[2][4], r3[2][8], r2[8];  // backward: B = W (K rows, N cols)
#pragma unroll
  for (int nt = 0; nt < 2; ++nt)
#pragma unroll
    for (int c = 0; c < 4; ++c) {
      int k = c * 4 + 2 * half;
      f2[nt][c].x = W2[(nt * 16 + q) * 16 + k];
      f2[nt][c].y = W2[(nt * 16 + q) * 16 + k + 1];
      r4[nt][c].x = W4[k * 32 + nt * 16 + q];
      r4[nt][c].y = W4[(k + 1) * 32 + nt * 16 + q];
    }
#pragma unroll
  for (int nt = 0; nt < 2; ++nt)
#pragma unroll
    for (int c = 0; c < 8; ++c) {
      int k = c * 4 + 2 * half;
      f3[nt][c].x = W3[(nt * 16 + q) * 32 + k];
      f3[nt][c].y = W3[(nt * 16 + q) * 32 + k + 1];
      r3[nt][c].x = W3[k * 32 + nt * 16 + q];
      r3[nt][c].y = W3[(k + 1) * 32 + nt * 16 + q];
    }
#pragma unroll
  for (int c = 0; c < 8; ++c) {
    int k = c * 4 + 2 * half;
    f4[c].x = W4[q * 32 + k];
    f4[c].y = W4[q * 32 + k + 1];
    r2[c].x = W2[k * 16 + q];
    r2[c].y = W2[(k + 1) * 16 + q];
  }

  const float bb1 = b1[q];
  const float bb2[2] = {b2[q], b2[16 + q]};
  const float bb3[2] = {b3[q], b3[16 + q]};
  const float bb4 = b4[q];
  const float bb5 = b5[0];
  const float w5q = W5[q];

  // ---------------- time stepping ------------------------------------------
  for (int i = 0; i < nsteps; ++i) {
    S.yb[2 * q + half] = yown;
    __syncthreads();

    // ---- forward L1: (16p x 2) @ W1^T -> h1 (16p x 16)
    {
      v2f a;
      a.x = half ? 0.f : S.yb[2 * q + 0];
      a.y = half ? 0.f : S.yb[2 * q + 1];
      v8f acc = {};
      acc = wmma4(a, f1, acc);
#pragma unroll
      for (int r = 0; r < 8; ++r)
        S.h1[(r + 8 * half) * 16 + q] = eluf(acc[r] + bb1);
    }
    __syncthreads();

    // ---- forward L2: h1 (K=16) -> h2 (16p x 32)
#pragma unroll
    for (int nt = 0; nt < 2; ++nt) {
      v8f acc = {};
#pragma unroll
      for (int c = 0; c < 4; ++c) acc = wmma4(afrag(S.h1, q, half, c, 16), f2[nt][c], acc);
#pragma unroll
      for (int r = 0; r < 8; ++r)
        S.h2[(r + 8 * half) * 32 + nt * 16 + q] = eluf(acc[r] + bb2[nt]);
    }
    __syncthreads();

    // ---- forward L3: h2 (K=32) -> h3 (16p x 32)
#pragma unroll
    for (int nt = 0; nt < 2; ++nt) {
      v8f acc = {};
#pragma unroll
      for (int c = 0; c < 8; ++c) acc = wmma4(afrag(S.h2, q, half, c, 32), f3[nt][c], acc);
#pragma unroll
      for (int r = 0; r < 8; ++r)
        S.h3[(r + 8 * half) * 32 + nt * 16 + q] = eluf(acc[r] + bb3[nt]);
    }
    __syncthreads();

    // ---- forward L4: h3 (K=32) -> h4 (16p x 16)
    {
      v8f acc = {};
#pragma unroll
      for (int c = 0; c < 8; ++c) acc = wmma4(afrag(S.h3, q, half, c, 32), f4[c], acc);
#pragma unroll
      for (int r = 0; r < 8; ++r)
        S.h4[(r + 8 * half) * 16 + q] = eluf(acc[r] + bb4);
    }
    __syncthreads();

    // ---- head: z5 = h4 @ W5^T + b5 ; g5 = sigmoid(z5) (softplus' for sum)
    if (half == 0) {
      float z = bb5;
#pragma unroll
      for (int f = 0; f < 16; ++f) z += S.h4[q * 16 + f] * sW5[f];
      S.g5[q] = 1.f / (1.f + __expf(-z));
    }
    __syncthreads();

    // ---- g4 = g5 * W5[f] * elu'(z4)  -> gA (16p x 16)
#pragma unroll
    for (int r = 0; r < 8; ++r) {
      int p = r + 8 * half;
      float hp = S.h4[p * 16 + q];
      S.gA[p * 16 + q] = S.g5[p] * w5q * elup_from_h(hp);
    }
    __syncthreads();

    // ---- backward L4: g3 = (g4 @ W4) * elu'(z3)  -> gB (16p x 32)
#pragma unroll
    for (int nt = 0; nt < 2; ++nt) {
      v8f acc = {};
#pragma unroll
      for (int c = 0; c < 4; ++c) acc = wmma4(afrag(S.gA, q, half, c, 16), r4[nt][c], acc);
#pragma unroll
      for (int r = 0; r < 8; ++r) {
        int p = r + 8 * half, f = nt * 16 + q;
        S.gB[p * 32 + f] = acc[r] * elup_from_h(S.h3[p * 32 + f]);
      }
    }
    __syncthreads();

    // ---- backward L3: g2 = (g3 @ W3) * elu'(z2)  -> gA (16p x 32)
#pragma unroll
    for (int nt = 0; nt < 2; ++nt) {
      v8f acc = {};
#pragma unroll
      for (int c = 0; c < 8; ++c) acc = wmma4(afrag(S.gB, q, half, c, 32), r3[nt][c], acc);
#pragma unroll
      for (int r = 0; r < 8; ++r) {
        int p = r + 8 * half, f = nt * 16 + q;
        S.gA[p * 32 + f] = acc[r] * elup_from_h(S.h2[p * 32 + f]);
      }
    }
    __syncthreads();

    // ---- backward L2: g1 = (g2 @ W2) * elu'(z1)  -> gB (16p x 16)
    {
      v8f acc = {};
#pragma unroll
      for (int c = 0; c < 8; ++c) acc = wmma4(afrag(S.gA, q, half, c, 32), r2[c], acc);
#pragma unroll
      for (int r = 0; r < 8; ++r) {
        int p = r + 8 * half;
        S.gB[p * 16 + q] = acc[r] * elup_from_h(S.h1[p * 16 + q]);
      }
    }
    __syncthreads();

    // ---- grad_y = g1 @ W1 (per-lane VALU), tilt, Euler-Maruyama update
    float gy = 0.f;
#pragma unroll
    for (int f = 0; f < 16; ++f) gy += S.gB[q * 16 + f] * sW1[f * 2 + half];

    float t = t0 + (float)i * dtv;
    float s0 = (t < tcrit) ? sp0 : ss0;
    float s1 = (t < tcrit) ? sp1 : ss1;
    float tilt = s0 * wt0 + s1 * wt1;
    float dw = sqdt * rng_normal((unsigned)i, gidx);
    yown += -(gy + tilt) * dtv + SIGMA_C * dw;
    __syncthreads();
  }

  out[(size_t)b * NP * 2 + (n0 + q) * 2 + half] = yown;
}

extern "C" void kernel_launch(void* const* d_in, const int* in_sizes, int n_in,
                              void* d_out, int out_size, void* d_ws,
                              size_t ws_size, hipStream_t stream) {
  (void)in_sizes; (void)n_in; (void)out_size; (void)d_ws; (void)ws_size;
  const float* x  = (const float*)d_in[0];
  const float* W1 = (const float*)d_in[1];
  const float* b1 = (const float*)d_in[2];
  const float* W2 = (const float*)d_in[3];
  const float* b2 = (const float*)d_in[4];
  const float* W3 = (const float*)d_in[5];
  const float* b3 = (const float*)d_in[6];
  const float* W4 = (const float*)d_in[7];
  const float* b4 = (const float*)d_in[8];
  const float* W5 = (const float*)d_in[9];
  const float* b5 = (const float*)d_in[10];
  const float* Wt = (const float*)d_in[11];
  const float* dt = (const float*)d_in[12];
  const int* ns   = (const int*)d_in[13];
  float* out = (float*)d_out;

  const int tiles = NB * NP / 16;          // 16384 particle tiles
  const int blocks = tiles / WAVES;        // 4096 blocks of 4 waves
  sde_mlp_kernel<<<blocks, WAVES * 32, 0, stream>>>(
      x, W1, b1, W2, b2, W3, b3, W4, b4, W5, b5, Wt, dt, ns, out);
}